// decoder_only_ssm_28673201668184
// MI455X (gfx1250) — compile-verified
//
#include <hip/hip_runtime.h>
#include <math.h>

// ---- problem constants (match reference) ----
#define BB   32
#define LL   2048
#define HH   8
#define LT   2055          // L + H - 1
#define DMET 16
#define SDIM 27
#define FDIM 44            // DMET + 1 + SDIM
#define DM   256
#define NL   4
#define DS   64
#define TE   256
#define NELEM ((size_t)BB * DM * LT)   // 16,834,560 floats

typedef __attribute__((ext_vector_type(2))) float v2f;
typedef __attribute__((ext_vector_type(8))) float v8f;

__device__ __forceinline__ float sigmoidf_(float x) { return 1.f / (1.f + expf(-x)); }
__device__ __forceinline__ float siluf_(float x)    { return x * sigmoidf_(x); }
__device__ __forceinline__ float gelu_tanh_(float x) {
    float x3 = x * x * x;
    return 0.5f * x * (1.f + tanhf(0.7978845608028654f * (x + 0.044715f * x3)));
}

// ---------------------------------------------------------------------------
// Kernel 0: MPFourier time features + time MLP  -> t_b [B, DM]
// ---------------------------------------------------------------------------
__global__ void time_mlp_kernel(const float* __restrict__ t,
                                const float* __restrict__ freqs,
                                const float* __restrict__ phases,
                                const float* __restrict__ W1, const float* __restrict__ b1,
                                const float* __restrict__ W2, const float* __restrict__ b2,
                                float* __restrict__ t_b) {
    __shared__ float tf[TE];
    __shared__ float h1[2 * TE];
    const int b = blockIdx.x;
    const int tid = threadIdx.x;
    const float tv = t[b];
    tf[tid] = cosf(tv * freqs[tid] + phases[tid]) * 1.4142135623730951f;
    __syncthreads();
    for (int j = tid; j < 2 * TE; j += blockDim.x) {
        float acc = b1[j];
        #pragma unroll 8
        for (int k = 0; k < TE; ++k) acc += tf[k] * W1[k * (2 * TE) + j];
        h1[j] = siluf_(acc);
    }
    __syncthreads();
    float acc = b2[tid];
    #pragma unroll 8
    for (int k = 0; k < 2 * TE; ++k) acc += h1[k] * W2[k * DM + tid];
    t_b[b * DM + tid] = acc;
}

// ---------------------------------------------------------------------------
// Kernel 1: feature fusion + input projection -> h [B, DM, LT]
// grid = (ceil(LT/16), B), block = 256 (thread == output channel d)
// ---------------------------------------------------------------------------
__global__ void input_proj_kernel(const float* __restrict__ x_past,
                                  const float* __restrict__ noisy_future,
                                  const float* __restrict__ x_future,
                                  const float* __restrict__ static_attr,
                                  const float* __restrict__ in_W,
                                  const float* __restrict__ in_b,
                                  const float* __restrict__ t_b,
                                  float* __restrict__ h) {
    __shared__ float sf[16 * FDIM];
    const int b  = blockIdx.y;
    const int l0 = blockIdx.x * 16;
    const int tid = threadIdx.x;

    for (int idx = tid; idx < 16 * FDIM; idx += blockDim.x) {
        const int ll = idx / FDIM;
        const int k  = idx - ll * FDIM;
        const int gl = l0 + ll;
        float v = 0.f;
        if (gl < LT) {
            if (k < DMET) {
                v = (gl < LL) ? x_past[((size_t)b * LL + gl) * DMET + k]
                              : x_future[((size_t)b * (HH - 1) + (gl - LL)) * DMET + k];
            } else if (k == DMET) {
                v = (gl >= LL - 1) ? noisy_future[(size_t)b * HH + (gl - (LL - 1))] : 0.f;
            } else {
                v = static_attr[(size_t)b * SDIM + (k - DMET - 1)];
            }
        }
        sf[idx] = v;
    }
    __syncthreads();

    const int d = tid;
    float wc[FDIM];
    #pragma unroll
    for (int k = 0; k < FDIM; ++k) wc[k] = in_W[k * DM + d];
    const float bias = in_b[d];
    const float tb   = t_b[b * DM + d];

    for (int ll = 0; ll < 16; ++ll) {
        const int gl = l0 + ll;
        if (gl < LT) {
            float acc = bias;
            #pragma unroll
            for (int k = 0; k < FDIM; ++k) acc += sf[ll * FDIM + k] * wc[k];
            if (gl >= LL - 1) acc += tb;
            h[((size_t)b * DM + d) * LT + gl] = acc;
        }
    }
}

// ---------------------------------------------------------------------------
// Kernel 2: S4D diagonal-state recurrence (== causal FFT conv in reference)
// One wave32 per (b,d); 2 complex states per lane (DS=64).
// y = gelu(ssm(u) + D*u), u = h.   grid = B*DM/8 blocks of 256.
// ---------------------------------------------------------------------------
__global__ void ssm_scan_kernel(int layer,
                                const float* __restrict__ log_dt,
                                const float* __restrict__ A_re, const float* __restrict__ A_im,
                                const float* __restrict__ C_re, const float* __restrict__ C_im,
                                const float* __restrict__ Dp,
                                const float* __restrict__ h,
                                float* __restrict__ y) {
    const int lane = threadIdx.x & 31;
    const int wid  = blockIdx.x * (blockDim.x >> 5) + (threadIdx.x >> 5);
    const int b = wid >> 8;        // / DM
    const int d = wid & (DM - 1);

    const size_t pbase = ((size_t)layer * DM + d) * DS;
    const float dt = expf(log_dt[layer * DM + d]);
    const float Dd = Dp[layer * DM + d];

    float wr[2], wi[2], cr[2], ci[2];
    #pragma unroll
    for (int s = 0; s < 2; ++s) {
        const int n = 2 * lane + s;
        const float ar = A_re[pbase + n], ai = A_im[pbase + n];
        const float Cr = C_re[pbase + n], Ci = C_im[pbase + n];
        const float dr = dt * ar, di = dt * ai;
        const float e = expf(dr);
        const float wrr = e * cosf(di), wii = e * sinf(di);   // w = exp(dt*A)
        wr[s] = wrr; wi[s] = wii;
        const float Er = wrr - 1.f, Ei = wii;                 // expm1(dt*A)
        const float den = ar * ar + ai * ai;
        const float qr = (Er * ar + Ei * ai) / den;           // expm1(dtA)/A
        const float qi = (Ei * ar - Er * ai) / den;
        cr[s] = Cr * qr - Ci * qi;                            // Ck
        ci[s] = Cr * qi + Ci * qr;
    }

    const float* up = h + ((size_t)b * DM + d) * LT;
    float*       yp = y + ((size_t)b * DM + d) * LT;

    float xr0 = 0.f, xi0 = 0.f, xr1 = 0.f, xi1 = 0.f;
    for (int l0 = 0; l0 < LT; l0 += 32) {
        const int li = l0 + lane;
        const float uch = (li < LT) ? up[li] : 0.f;
        float ymine = 0.f;
        for (int s = 0; s < 32; ++s) {
            const float u = __shfl(uch, s, 32);
            float t0r = wr[0] * xr0 - wi[0] * xi0 + u;
            float t0i = wr[0] * xi0 + wi[0] * xr0;
            xr0 = t0r; xi0 = t0i;
            float t1r = wr[1] * xr1 - wi[1] * xi1 + u;
            float t1i = wr[1] * xi1 + wi[1] * xr1;
            xr1 = t1r; xi1 = t1i;
            float p = cr[0] * xr0 - ci[0] * xi0 + cr[1] * xr1 - ci[1] * xi1;
            p += __shfl_xor(p, 1, 32);
            p += __shfl_xor(p, 2, 32);
            p += __shfl_xor(p, 4, 32);
            p += __shfl_xor(p, 8, 32);
            p += __shfl_xor(p, 16, 32);
            if (lane == s) ymine = 2.f * p;
        }
        if (li < LT) yp[li] = gelu_tanh_(ymine + Dd * uch);
    }
}

// ---------------------------------------------------------------------------
// Kernel 3: z = y^T @ out_W + out_b ; GLU ; residual into h (in place).
// V_WMMA_F32_16X16X4_F32, M=e-channel, N=timestep, K=d (64 steps of 4).
// Each wave computes the (e, e+256) tile pair -> GLU fused, no z buffer.
// grid = (ceil(LT/16), 2, B), block = 256 (8 waves), LDS = 16 KB y-tile.
// ---------------------------------------------------------------------------
__global__ void glu_gemm_kernel(int layer,
                                const float* __restrict__ out_W,
                                const float* __restrict__ out_b,
                                const float* __restrict__ y,
                                float* __restrict__ h) {
    __shared__ float ytile[DM * 16];
    const int b  = blockIdx.z;
    const int l0 = blockIdx.x * 16;
    const int tid = threadIdx.x;
    const int lane = tid & 31;
    const int wv   = tid >> 5;
    const int hf   = lane >> 4;    // lane half
    const int mn   = lane & 15;    // M-row (for A) / N-col (for B, C, D)

    // cooperative y-tile load: thread t owns channel row t, 16 timesteps
    {
        const float* yp = y + ((size_t)b * DM + tid) * LT;
        #pragma unroll
        for (int j = 0; j < 16; ++j) {
            const int gl = l0 + j;
            ytile[tid * 16 + j] = (gl < LT) ? yp[gl] : 0.f;
        }
        // hint the next l-tile into cache
        if (l0 + 16 < LT) __builtin_prefetch(yp + l0 + 16, 0, 1);
    }
    __syncthreads();

    const int e0 = (blockIdx.y * 8 + wv) * 16;     // z1 channel tile base, [0,256)
    const float* Wp = out_W + (size_t)layer * DM * (2 * DM);

    v8f acc1 = {}, acc2 = {};
    for (int d0 = 0; d0 < DM; d0 += 4) {
        const int r0 = d0 + 2 * hf;
        v2f a1, a2, bf;
        a1.x = Wp[(size_t)r0 * (2 * DM) + e0 + mn];
        a1.y = Wp[(size_t)(r0 + 1) * (2 * DM) + e0 + mn];
        a2.x = Wp[(size_t)r0 * (2 * DM) + e0 + DM + mn];
        a2.y = Wp[(size_t)(r0 + 1) * (2 * DM) + e0 + DM + mn];
        bf.x = ytile[r0 * 16 + mn];
        bf.y = ytile[(r0 + 1) * 16 + mn];
        acc1 = __builtin_amdgcn_wmma_f32_16x16x4_f32(false, a1, false, bf,
                                                     (short)0, acc1, false, false);
        acc2 = __builtin_amdgcn_wmma_f32_16x16x4_f32(false, a2, false, bf,
                                                     (short)0, acc2, false, false);
    }

    // epilogue: GLU + residual, in place on h
    const int l = l0 + mn;
    if (l < LT) {
        #pragma unroll
        for (int r = 0; r < 8; ++r) {
            const int ch = e0 + r + 8 * hf;
            const float z1 = acc1[r] + out_b[layer * (2 * DM) + ch];
            const float z2 = acc2[r] + out_b[layer * (2 * DM) + DM + ch];
            const size_t idx = ((size_t)b * DM + ch) * LT + l;
            h[idx] = h[idx] + z1 * sigmoidf_(z2);
        }
    }
}

// ---------------------------------------------------------------------------
// Kernel 4: deterministic per-channel BN statistics (block per channel)
// ---------------------------------------------------------------------------
__global__ void bn_stats_kernel(const float* __restrict__ h, float* __restrict__ stats) {
    __shared__ float ssum[256];
    __shared__ float ssq[256];
    const int d = blockIdx.x;
    const int tid = threadIdx.x;
    float s = 0.f, q = 0.f;
    const int total = BB * LT;
    for (int idx = tid; idx < total; idx += blockDim.x) {
        const int b = idx / LT;
        const int l = idx - b * LT;
        const float v = h[((size_t)b * DM + d) * LT + l];
        s += v; q += v * v;
    }
    ssum[tid] = s; ssq[tid] = q;
    __syncthreads();
    for (int off = 128; off > 0; off >>= 1) {
        if (tid < off) { ssum[tid] += ssum[tid + off]; ssq[tid] += ssq[tid + off]; }
        __syncthreads();
    }
    if (tid == 0) { stats[d] = ssum[0]; stats[DM + d] = ssq[0]; }
}

// ---------------------------------------------------------------------------
// Kernel 5: BN apply (in place)
// ---------------------------------------------------------------------------
__global__ void bn_apply_kernel(int layer,
                                const float* __restrict__ stats,
                                const float* __restrict__ gamma,
                                const float* __restrict__ beta,
                                float* __restrict__ h) {
    const size_t idx = (size_t)blockIdx.x * blockDim.x + threadIdx.x;
    if (idx >= NELEM) return;
    const int d = (int)((idx / LT) % DM);
    const float invN = 1.f / (float)(BB * LT);
    const float mean = stats[d] * invN;
    const float var  = stats[DM + d] * invN - mean * mean;
    const float g = gamma[layer * DM + d], be = beta[layer * DM + d];
    h[idx] = (h[idx] - mean) * rsqrtf(var + 1e-5f) * g + be;
}

// ---------------------------------------------------------------------------
// Kernel 6: head over the last H positions only
// ---------------------------------------------------------------------------
__global__ void head_kernel(const float* __restrict__ h,
                            const float* __restrict__ W1, const float* __restrict__ b1,
                            const float* __restrict__ W2, const float* __restrict__ b2,
                            float* __restrict__ out) {
    __shared__ float hv[DM];
    __shared__ float red[DM / 2];
    const int b = blockIdx.x;
    const int tid = threadIdx.x;
    for (int p = 0; p < HH; ++p) {
        const int l = LT - HH + p;
        hv[tid] = h[((size_t)b * DM + tid) * LT + l];
        __syncthreads();
        if (tid < DM / 2) {
            float acc = b1[tid];
            #pragma unroll 8
            for (int k = 0; k < DM; ++k) acc += hv[k] * W1[k * (DM / 2) + tid];
            red[tid] = siluf_(acc) * W2[tid];
        }
        __syncthreads();
        for (int off = 64; off > 0; off >>= 1) {
            if (tid < off) red[tid] += red[tid + off];
            __syncthreads();
        }
        if (tid == 0) out[b * HH + p] = red[0] + b2[0];
        __syncthreads();
    }
}

// ---------------------------------------------------------------------------
extern "C" void kernel_launch(void* const* d_in, const int* in_sizes, int n_in,
                              void* d_out, int out_size, void* d_ws, size_t ws_size,
                              hipStream_t stream) {
    const float* x_past     = (const float*)d_in[0];
    const float* noisy_fut  = (const float*)d_in[1];
    const float* t          = (const float*)d_in[2];
    const float* x_future   = (const float*)d_in[3];
    const float* static_at  = (const float*)d_in[4];
    const float* freqs      = (const float*)d_in[5];
    const float* phases     = (const float*)d_in[6];
    const float* in_W       = (const float*)d_in[7];
    const float* in_b       = (const float*)d_in[8];
    const float* tm_W1      = (const float*)d_in[9];
    const float* tm_b1      = (const float*)d_in[10];
    const float* tm_W2      = (const float*)d_in[11];
    const float* tm_b2      = (const float*)d_in[12];
    const float* log_dt     = (const float*)d_in[13];
    const float* A_re       = (const float*)d_in[14];
    const float* A_im       = (const float*)d_in[15];
    const float* C_re       = (const float*)d_in[16];
    const float* C_im       = (const float*)d_in[17];
    const float* Dp         = (const float*)d_in[18];
    const float* out_W      = (const float*)d_in[19];
    const float* out_b      = (const float*)d_in[20];
    const float* bn_gamma   = (const float*)d_in[21];
    const float* bn_beta    = (const float*)d_in[22];
    const float* head_W1    = (const float*)d_in[23];
    const float* head_b1    = (const float*)d_in[24];
    const float* head_W2    = (const float*)d_in[25];
    const float* head_b2    = (const float*)d_in[26];
    float* out = (float*)d_out;

    // workspace layout (floats): t_b | h | y | stats
    float* t_b  = (float*)d_ws;
    float* hbuf = t_b + BB * DM;
    float* ybuf = hbuf + NELEM;
    float* stats = ybuf + NELEM;

    const int LTILES = (LT + 15) / 16;   // 129

    time_mlp_kernel<<<BB, 256, 0, stream>>>(t, freqs, phases, tm_W1, tm_b1, tm_W2, tm_b2, t_b);

    input_proj_kernel<<<dim3(LTILES, BB), 256, 0, stream>>>(
        x_past, noisy_fut, x_future, static_at, in_W, in_b, t_b, hbuf);

    for (int i = 0; i < NL; ++i) {
        ssm_scan_kernel<<<(BB * DM) / 8, 256, 0, stream>>>(
            i, log_dt, A_re, A_im, C_re, C_im, Dp, hbuf, ybuf);
        glu_gemm_kernel<<<dim3(LTILES, 2, BB), 256, 0, stream>>>(
            i, out_W, out_b, ybuf, hbuf);
        bn_stats_kernel<<<DM, 256, 0, stream>>>(hbuf, stats);
        bn_apply_kernel<<<(unsigned)((NELEM + 255) / 256), 256, 0, stream>>>(
            i, stats, bn_gamma, bn_beta, hbuf);
    }

    head_kernel<<<BB, 256, 0, stream>>>(hbuf, head_W1, head_b1, head_W2, head_b2, out);
}